// BlockGNN_85469849190401
// MI455X (gfx1250) — compile-verified
//
#include <hip/hip_runtime.h>
#include <hip/hip_bf16.h>

typedef float v2f __attribute__((ext_vector_type(2)));
typedef float v8f __attribute__((ext_vector_type(8)));

#define HDIM 128

// ---------------- utility ----------------
__global__ void k_zero(float* __restrict__ p, long long n) {
  long long i = (long long)blockIdx.x * blockDim.x + threadIdx.x;
  if (i < n) p[i] = 0.0f;
}

// ---------------- degree ----------------
__global__ void k_degree(const int* __restrict__ dst, float* __restrict__ deg, int E) {
  int e = blockIdx.x * blockDim.x + threadIdx.x;
  if (e < E) atomicAdd(&deg[dst[e]], 1.0f);
}

__global__ void k_dinv(float* __restrict__ deg_inout, int N) {
  int i = blockIdx.x * blockDim.x + threadIdx.x;
  if (i < N) deg_inout[i] = rsqrtf(deg_inout[i] + 1.0f);  // deg^{-1/2}, self-loop included
}

// ---------------- fp32 WMMA GEMM: out[N,128] = A[N,128] @ W[128,128] ----------------
// Block = 256 threads (8 waves), covers 128 rows. Each wave owns ONE 16-row tile and
// ALL 128 output columns (8 x v8f accumulators -> 8 independent WMMA chains).
// W staged once per block in LDS (64 KB); A rows read exactly once from global.
__global__ void k_gemm_wmma(const float* __restrict__ A, const float* __restrict__ W,
                            float* __restrict__ out, int nrows) {
  __shared__ float sW[HDIM * HDIM];  // 64 KB
  const int tid = threadIdx.x;

  // cooperative float4 load of W into LDS: 4096 float4 / 256 threads = 16 each
  const float4* Wv = (const float4*)W;
  float4* sWv = (float4*)sW;
#pragma unroll
  for (int i = 0; i < 16; ++i) sWv[tid + 256 * i] = Wv[tid + 256 * i];
  __syncthreads();

  const int wave = tid >> 5;         // 0..7 -> row tile within block
  const int lane = tid & 31;
  const int lm   = lane & 15;        // M (for A) / N (for B)
  const int kk   = (lane >> 4) * 2;  // lanes 0-15: K offs {0,1}; lanes 16-31: {2,3}
  const int m0   = blockIdx.x * 128 + wave * 16;
  if (m0 >= nrows) return;           // whole tile out of range (after staging+sync)

  int row = m0 + lm;
  if (row >= nrows) row = nrows - 1; // clamp (stores are guarded)
  const float* __restrict__ arow = A + (size_t)row * HDIM;

  v8f acc[8];
#pragma unroll
  for (int n = 0; n < 8; ++n) acc[n] = (v8f){};

  for (int k = 0; k < HDIM; k += 4) {
    v2f a;
    // A 16x4 fragment (MxK): v0 = K=k+kk, v1 = K=k+kk+1, M = lane&15
    a[0] = arow[k + kk];
    a[1] = arow[k + kk + 1];
    const float* __restrict__ wrow0 = sW + (k + kk) * HDIM + lm;
    const float* __restrict__ wrow1 = wrow0 + HDIM;
#pragma unroll
    for (int n = 0; n < 8; ++n) {
      v2f b;
      // B 4x16 fragment (KxN): v0 row K=k+kk, v1 row K=k+kk+1, N = n*16 + (lane&15)
      b[0] = wrow0[n * 16];
      b[1] = wrow1[n * 16];
      acc[n] = __builtin_amdgcn_wmma_f32_16x16x4_f32(false, a, false, b,
                                                     (short)0, acc[n], false, false);
    }
  }

  // C/D 16x16 f32 layout: VGPR r -> lanes 0-15: M=r, N=lane; lanes 16-31: M=r+8, N=lane-16
  const int mbase = m0 + ((lane >> 4) ? 8 : 0);
#pragma unroll
  for (int n = 0; n < 8; ++n) {
    const int col = n * 16 + lm;
#pragma unroll
    for (int r = 0; r < 8; ++r) {
      int m = mbase + r;
      if (m < nrows) out[(size_t)m * HDIM + col] = acc[n][r];
    }
  }
}

// ---------------- edge scatter: agg[dst] += xw[src] * dinv[src]*dinv[dst] ----------------
// 32 consecutive threads per edge; float4 channel groups (coalesced 512B reads per edge).
__global__ void k_scatter(const float* __restrict__ xw, const int* __restrict__ src,
                          const int* __restrict__ dst, const float* __restrict__ dinv,
                          float* __restrict__ agg, long long nwork) {
  long long g = (long long)blockIdx.x * blockDim.x + threadIdx.x;
  if (g >= nwork) return;
  const int e  = (int)(g >> 5);
  const int c4 = ((int)g & 31) * 4;
  const int s = src[e], d = dst[e];
  const float coef = dinv[s] * dinv[d];
  const float4 v = *(const float4*)(xw + (size_t)s * HDIM + c4);
  float* ap = agg + (size_t)d * HDIM + c4;
  atomicAdd(ap + 0, v.x * coef);
  atomicAdd(ap + 1, v.y * coef);
  atomicAdd(ap + 2, v.z * coef);
  atomicAdd(ap + 3, v.w * coef);
}

// ---------------- combine: out = relu(agg + xw * dinv^2 + b) (out may alias xw) ----------
__global__ void k_combine(const float* __restrict__ agg, const float* __restrict__ xw,
                          const float* __restrict__ dinv, const float* __restrict__ bias,
                          float* __restrict__ out, long long n) {
  long long i = (long long)blockIdx.x * blockDim.x + threadIdx.x;
  if (i >= n) return;
  const int node = (int)(i >> 7);
  const int c    = (int)(i & (HDIM - 1));
  const float di = dinv[node];
  const float v  = agg[i] + xw[i] * (di * di) + bias[c];
  out[i] = fmaxf(v, 0.0f);
}

// ---------------- pooling ----------------
__global__ void k_count(const int* __restrict__ batch, float* __restrict__ cnt, int N) {
  int i = blockIdx.x * blockDim.x + threadIdx.x;
  if (i < N) atomicAdd(&cnt[batch[i]], 1.0f);
}

__global__ void k_pool(const float* __restrict__ h, const int* __restrict__ batch,
                       float* __restrict__ pooled, long long nwork) {
  long long g = (long long)blockIdx.x * blockDim.x + threadIdx.x;
  if (g >= nwork) return;
  const int n  = (int)(g >> 5);
  const int c4 = ((int)g & 31) * 4;
  const int b  = batch[n];
  const float4 v = *(const float4*)(h + (size_t)n * HDIM + c4);
  float* pp = pooled + (size_t)b * HDIM + c4;
  atomicAdd(pp + 0, v.x);
  atomicAdd(pp + 1, v.y);
  atomicAdd(pp + 2, v.z);
  atomicAdd(pp + 3, v.w);
}

// ---------------- final linear: out[G,C] = (pooled/cnt) @ Wlin + blin ----------------
__global__ void k_final(const float* __restrict__ pooled, const float* __restrict__ cnt,
                        const float* __restrict__ Wlin, const float* __restrict__ blin,
                        float* __restrict__ out, int G, int C) {
  int i = blockIdx.x * blockDim.x + threadIdx.x;
  if (i >= G * C) return;
  const int g = i / C, cc = i % C;
  const float* pr = pooled + (size_t)g * HDIM;
  float sum = 0.0f;
#pragma unroll 8
  for (int k = 0; k < HDIM; ++k) sum += pr[k] * Wlin[k * C + cc];
  out[i] = sum / fmaxf(cnt[g], 1.0f) + blin[cc];
}

// =====================================================================================
extern "C" void kernel_launch(void* const* d_in, const int* in_sizes, int n_in,
                              void* d_out, int out_size, void* d_ws, size_t ws_size,
                              hipStream_t stream) {
  const float* x     = (const float*)d_in[0];
  const int*   ei    = (const int*)  d_in[1];
  const int*   batch = (const int*)  d_in[2];
  const float* W0    = (const float*)d_in[3];
  const float* b0    = (const float*)d_in[4];
  const float* W1    = (const float*)d_in[5];
  const float* b1    = (const float*)d_in[6];
  const float* W2    = (const float*)d_in[7];
  const float* b2    = (const float*)d_in[8];
  const float* Wlin  = (const float*)d_in[9];
  const float* blin  = (const float*)d_in[10];
  float* out = (float*)d_out;

  const int N = in_sizes[2];
  const int E = in_sizes[1] / 2;
  const int C = in_sizes[10];
  const int G = out_size / C;
  const int* src = ei;
  const int* dst = ei + E;

  // workspace layout (floats)
  float* ws     = (float*)d_ws;
  float* dinv   = ws;                              // N    (deg, then dinv in-place)
  float* B1     = dinv + N;                        // N*H
  float* B2     = B1 + (size_t)N * HDIM;           // N*H
  float* agg    = B2 + (size_t)N * HDIM;           // N*H
  float* pooled = agg + (size_t)N * HDIM;          // G*H
  float* cnt    = pooled + (size_t)G * HDIM;       // G

  const long long NH = (long long)N * HDIM;
  const long long escatter = (long long)E * 32;
  const long long npool = (long long)N * 32;

  const int T = 256;
  dim3 blk(T);
  auto nb = [](long long n, int t) { return (unsigned)((n + t - 1) / t); };
  const unsigned gemm_blocks = (unsigned)((N + 127) / 128);

  // degrees -> dinv (edge-structure invariant across layers)
  k_zero  <<<nb(N, T), blk, 0, stream>>>(dinv, N);
  k_degree<<<nb(E, T), blk, 0, stream>>>(dst, dinv, E);
  k_dinv  <<<nb(N, T), blk, 0, stream>>>(dinv, N);

  // ---- layer 0: x -> B1 ----
  k_gemm_wmma<<<dim3(gemm_blocks), blk, 0, stream>>>(x, W0, B1, N);
  k_zero     <<<nb(NH, T), blk, 0, stream>>>(agg, NH);
  k_scatter  <<<nb(escatter, T), blk, 0, stream>>>(B1, src, dst, dinv, agg, escatter);
  k_combine  <<<nb(NH, T), blk, 0, stream>>>(agg, B1, dinv, b0, B1, NH);

  // ---- layer 1: B1 -> B2 ----
  k_gemm_wmma<<<dim3(gemm_blocks), blk, 0, stream>>>(B1, W1, B2, N);
  k_zero     <<<nb(NH, T), blk, 0, stream>>>(agg, NH);
  k_scatter  <<<nb(escatter, T), blk, 0, stream>>>(B2, src, dst, dinv, agg, escatter);
  k_combine  <<<nb(NH, T), blk, 0, stream>>>(agg, B2, dinv, b1, B2, NH);

  // ---- layer 2: B2 -> B1 ----
  k_gemm_wmma<<<dim3(gemm_blocks), blk, 0, stream>>>(B2, W2, B1, N);
  k_zero     <<<nb(NH, T), blk, 0, stream>>>(agg, NH);
  k_scatter  <<<nb(escatter, T), blk, 0, stream>>>(B1, src, dst, dinv, agg, escatter);
  k_combine  <<<nb(NH, T), blk, 0, stream>>>(agg, B1, dinv, b2, B1, NH);

  // ---- pooling + final linear ----
  k_zero <<<nb((long long)G * HDIM, T), blk, 0, stream>>>(pooled, (long long)G * HDIM);
  k_zero <<<nb(G, T), blk, 0, stream>>>(cnt, G);
  k_count<<<nb(N, T), blk, 0, stream>>>(batch, cnt, N);
  k_pool <<<nb(npool, T), blk, 0, stream>>>(B1, batch, pooled, npool);
  k_final<<<nb((long long)G * C, T), blk, 0, stream>>>(pooled, cnt, Wlin, blin, out, G, C);
}